// DirectImageAlign_42863773614740
// MI455X (gfx1250) — compile-verified
//
#include <hip/hip_runtime.h>
#include <math.h>

namespace {
constexpr int   kB     = 4;
constexpr int   kC     = 16;
constexpr int   kH     = 240;
constexpr int   kW     = 320;
constexpr int   kHW    = kH * kW;
constexpr int   kNB    = 60;     // blocks per batch (60*256 = 15360 -> exactly 5 pixels/thread)
constexpr int   kBlock = 256;    // 8 wave32
constexpr int   kWaves = kBlock / 32;
constexpr float kLam   = 0.01f;
constexpr float kHuber = 1.345f;
constexpr float kSqrtHuber = 1.1597414f;   // sqrt(1.345)
constexpr float kLmEps = 1e-6f;
constexpr int   kNIter = 5;
}

typedef float v2f __attribute__((ext_vector_type(2)));
typedef float v8f __attribute__((ext_vector_type(8)));

__device__ __forceinline__ int iclampi(int x, int lo, int hi) {
  return x < lo ? lo : (x > hi ? hi : x);
}

struct Samp { float val, gx, gy; };

// Bilinear sample of `plane` (H*W, row-major) and of its edge-clamped central-
// difference gradients at the 4 corners, with zero-outside corner masking
// already folded into w00..w11.  12 unique loads.
__device__ __forceinline__ Samp sample3(const float* __restrict__ plane,
                                        int xm1, int x0c, int x1c, int x2c,
                                        int ym1, int y0c, int y1c, int y2c,
                                        float w00, float w01, float w10, float w11) {
  const float* rm = plane + ym1 * kW;
  const float* r0 = plane + y0c * kW;
  const float* r1 = plane + y1c * kW;
  const float* r2 = plane + y2c * kW;
  float A  = r0[x0c], Bv = r0[x1c], Cv = r1[x0c], D  = r1[x1c];
  float L0 = r0[xm1], R0 = r0[x2c], L1 = r1[xm1], R1 = r1[x2c];
  float T0 = rm[x0c], T1 = rm[x1c], B0 = r2[x0c], B1 = r2[x1c];
  Samp s;
  s.val = w00 * A + w01 * Bv + w10 * Cv + w11 * D;
  s.gx  = 0.5f * (w00 * (Bv - L0) + w01 * (R0 - A) + w10 * (D - L1) + w11 * (R1 - Cv));
  s.gy  = 0.5f * (w00 * (Cv - T0) + w01 * (D - T1) + w10 * (B0 - A) + w11 * (B1 - Bv));
  return s;
}

// ---------------------------------------------------------------------------
// Kernel A: per-pixel residual rows -> WMMA-accumulated 7x7 normal-equation
// block per wave -> deterministic block partial (49 floats) into d_ws.
// ---------------------------------------------------------------------------
__global__ __launch_bounds__(kBlock) void dia_reduce(
    const float* __restrict__ pose,     // B x 16 (current estimate, lives in d_out)
    const float* __restrict__ I0,       // B x C x H x W
    const float* __restrict__ I1,       // B x C x H x W
    const float* __restrict__ intr,     // B x 4
    const float* __restrict__ depth0,   // B x 1 x H x W
    const float* __restrict__ depth1,   // B x 1 x H x W
    float* __restrict__ partial)        // B x kNB x 49
{
  // Wave-private WMMA staging: 16 K-slices x 16 lanes of packed rows
  // slice k holds rows (2k) in slots 0..6 and (2k+1) in slots 8..14.
  __shared__ float S[kWaves][16][16];
  __shared__ float red[kWaves][49];

  const int b     = blockIdx.x / kNB;
  const int blk   = blockIdx.x % kNB;
  const int tid   = threadIdx.x;
  const int wave  = tid >> 5;
  const int lane  = tid & 31;
  const int laneLo = lane & 15;
  const int hp    = (lane >= 16) ? 2 : 0;

  // zero the staging buffer once (covers the always-zero pad slots 7 and 15)
  {
    float* sz = &S[wave][0][0] + lane * 8;
    #pragma unroll
    for (int j = 0; j < 8; ++j) sz[j] = 0.0f;
  }

  const float* P = pose + b * 16;
  const float R00 = P[0], R01 = P[1],  R02 = P[2],  tx = P[3];
  const float R10 = P[4], R11 = P[5],  R12 = P[6],  ty = P[7];
  const float R20 = P[8], R21 = P[9],  R22 = P[10], tz = P[11];
  const float fx = intr[b * 4 + 0], fy = intr[b * 4 + 1];
  const float cx = intr[b * 4 + 2], cy = intr[b * 4 + 3];
  const float invFx = 1.0f / fx, invFy = 1.0f / fy;   // hoisted exact divides

  const float* I0b = I0 + (size_t)b * kC * kHW;
  const float* I1b = I1 + (size_t)b * kC * kHW;
  const float* d0b = depth0 + (size_t)b * kHW;
  const float* d1b = depth1 + (size_t)b * kHW;

  v8f acc = {0.f, 0.f, 0.f, 0.f, 0.f, 0.f, 0.f, 0.f};

  for (int base = blk * kBlock + tid; base < kHW; base += kNB * kBlock) {
    const int ph = base / kW;
    const int pw = base - ph * kW;

    const float d1  = d1b[base];
    const float X1x = d1 * ((float)pw - cx) * invFx;
    const float X1y = d1 * ((float)ph - cy) * invFy;
    const float X1z = d1;

    const float x = R00 * X1x + R01 * X1y + R02 * X1z + tx;
    const float y = R10 * X1x + R11 * X1y + R12 * X1z + ty;
    const float z = R20 * X1x + R21 * X1y + R22 * X1z + tz;

    const float vproj = (z > 1e-6f) ? 1.f : 0.f;
    const float zs    = (z > 1e-6f) ? z : 1.0f;
    const float invz  = __builtin_amdgcn_rcpf(zs);     // v_rcp_f32
    const float u0 = fx * x * invz + cx;
    const float v0 = fy * y * invz + cy;

    const float dref0 = d0b[base];
    const float inb = ((u0 > 0.f) && (u0 < (float)(kW - 1)) &&
                       (v0 > 0.f) && (v0 < (float)(kH - 1))) ? 1.f : 0.f;
    const float validf = inb * vproj * ((dref0 > 0.f) ? 1.f : 0.f) * ((d1 > 0.f) ? 1.f : 0.f);

    // Jw = Jp * [I3 | -skew(X0)]  (2x6), Jt row2 for the geometric residual
    const float aT = fx * invz, cT = -fx * x * invz * invz;
    const float eT = fy * invz, fT = -fy * y * invz * invz;
    const float Jw0[6] = { aT, 0.f, cT,  cT * y,      aT * z - cT * x, -aT * y };
    const float Jw1[6] = { 0.f, eT, fT, -eT * z + fT * y, -fT * x,      eT * x };
    const float Jt2[6] = { 0.f, 0.f, 1.f, y, -x, 0.f };

    // grid_sample normalize -> clip(+-2) -> denormalize collapses to a clamp
    const float xs = fminf(fmaxf(u0, -0.5f * (float)(kW - 1)), 1.5f * (float)(kW - 1));
    const float ys = fminf(fmaxf(v0, -0.5f * (float)(kH - 1)), 1.5f * (float)(kH - 1));
    const float x0f = floorf(xs), y0f = floorf(ys);
    const float fx1 = xs - x0f, fy1 = ys - y0f;
    const int x0i = (int)x0f, y0i = (int)y0f;

    const float i00 = (x0i >= 0 && x0i <= kW - 1 && y0i >= 0 && y0i <= kH - 1) ? 1.f : 0.f;
    const float i01 = (x0i + 1 >= 0 && x0i + 1 <= kW - 1 && y0i >= 0 && y0i <= kH - 1) ? 1.f : 0.f;
    const float i10 = (x0i >= 0 && x0i <= kW - 1 && y0i + 1 >= 0 && y0i + 1 <= kH - 1) ? 1.f : 0.f;
    const float i11 = (x0i + 1 >= 0 && x0i + 1 <= kW - 1 && y0i + 1 >= 0 && y0i + 1 <= kH - 1) ? 1.f : 0.f;
    const float w00 = (1.f - fx1) * (1.f - fy1) * i00;
    const float w01 = fx1 * (1.f - fy1) * i01;
    const float w10 = (1.f - fx1) * fy1 * i10;
    const float w11 = fx1 * fy1 * i11;

    const int x0c = iclampi(x0i, 0, kW - 1), x1c = iclampi(x0i + 1, 0, kW - 1);
    const int xm1 = iclampi(x0i - 1, 0, kW - 1), x2c = iclampi(x0i + 2, 0, kW - 1);
    const int y0c = iclampi(y0i, 0, kH - 1), y1c = iclampi(y0i + 1, 0, kH - 1);
    const int ym1 = iclampi(y0i - 1, 0, kH - 1), y2c = iclampi(y0i + 2, 0, kH - 1);

    float* rp = &S[wave][lane >> 1][(lane & 1) * 8];

    for (int c = 0; c < kC + 1; ++c) {   // 16 photometric rows + 1 geometric row
      float jrow[6];
      float res;
      if (c < kC) {
        const float* plane = I0b + (size_t)c * kHW;
        Samp s = sample3(plane, xm1, x0c, x1c, x2c, ym1, y0c, y1c, y2c, w00, w01, w10, w11);
        res = I1b[(size_t)c * kHW + base] - s.val;
        #pragma unroll
        for (int k = 0; k < 6; ++k) jrow[k] = s.gx * Jw0[k] + s.gy * Jw1[k];
      } else {
        Samp s = sample3(d0b, xm1, x0c, x1c, x2c, ym1, y0c, y1c, y2c, w00, w01, w10, w11);
        res = kLam * (z - s.val);
        #pragma unroll
        for (int k = 0; k < 6; ++k) jrow[k] = kLam * (s.gx * Jw0[k] + s.gy * Jw1[k] - Jt2[k]);
      }
      res *= validf;
      #pragma unroll
      for (int k = 0; k < 6; ++k) jrow[k] *= validf;

      // Huber: sqrt(delta/ar) = sqrt(delta) * rsqrt(ar)  (ar >= delta > 0 here)
      const float ar = fabsf(res);
      const float sw = (ar <= kHuber) ? 1.f : kSqrtHuber * __builtin_amdgcn_rsqf(ar);

      // stage u = [sw*j0..j5, sw*r] into the packed WMMA operand layout
      ((float4*)rp)[0] = make_float4(sw * jrow[0], sw * jrow[1], sw * jrow[2], sw * jrow[3]);
      ((float2*)rp)[2] = make_float2(sw * jrow[4], sw * jrow[5]);
      rp[6] = sw * res;

      // 4 WMMAs consume all 32 lane-rows (2 rows per K-slice).  A and B
      // operand layouts mirror, so the same register pair feeds both:
      // C[m][n] += sum_k v_k[m] * v_k[n]  ->  top-left 7x7 = even-lane JtWJ | JtWr (odd lanes: rows/cols 8..14).
      #pragma unroll
      for (int kb = 0; kb < 16; kb += 4) {
        v2f av;
        av.x = S[wave][kb + hp][laneLo];
        av.y = S[wave][kb + hp + 1][laneLo];
        acc = __builtin_amdgcn_wmma_f32_16x16x4_f32(
            /*neg_a=*/false, av, /*neg_b=*/false, av,
            /*c_mod=*/(short)0, acc, /*reuse_a=*/false, /*reuse_b=*/false);
      }
    }
  }

  // C layout: VGPR r, lanes 0..15 -> (M=r, N=lane).  7x7 block lives in
  // lanes 0..6, VGPRs 0..6.
  if (lane < 7) {
    #pragma unroll
    for (int r = 0; r < 7; ++r) red[wave][r * 7 + lane] = acc[r];
  }
  __syncthreads();
  if (tid < 49) {
    float s = 0.f;
    #pragma unroll
    for (int wv = 0; wv < kWaves; ++wv) s += red[wv][tid];
    partial[((size_t)b * kNB + blk) * 49 + tid] = s;
  }
}

// ---------------------------------------------------------------------------
// Kernel B: deterministic partial reduction, damped 6x6 solve, se3_exp, pose
// update (in place in d_out).  One block per batch.
// ---------------------------------------------------------------------------
__global__ __launch_bounds__(64) void dia_solve(const float* __restrict__ partial,
                                                float* __restrict__ pose) {
  const int b = blockIdx.x;
  const int tid = threadIdx.x;
  __shared__ float M[49];
  if (tid < 49) {
    float s = 0.f;
    for (int blk = 0; blk < kNB; ++blk)
      s += partial[((size_t)b * kNB + blk) * 49 + tid];
    M[tid] = s;
  }
  __syncthreads();
  if (tid != 0) return;

  // augmented system [H | g]
  float A[6][7];
  for (int m = 0; m < 6; ++m) {
    for (int n = 0; n < 6; ++n) A[m][n] = M[m * 7 + n] + (m == n ? kLmEps : 0.f);
    A[m][6] = M[m * 7 + 6];
  }
  // Gaussian elimination with partial pivoting
  for (int col = 0; col < 6; ++col) {
    int piv = col;
    float best = fabsf(A[col][col]);
    for (int r = col + 1; r < 6; ++r) {
      float v = fabsf(A[r][col]);
      if (v > best) { best = v; piv = r; }
    }
    if (piv != col)
      for (int k = 0; k < 7; ++k) { float t = A[col][k]; A[col][k] = A[piv][k]; A[piv][k] = t; }
    const float inv = 1.f / A[col][col];
    for (int r = col + 1; r < 6; ++r) {
      const float f = A[r][col] * inv;
      for (int k = col; k < 7; ++k) A[r][k] -= f * A[col][k];
    }
  }
  float xi[6];
  for (int r = 5; r >= 0; --r) {
    float s = A[r][6];
    for (int k = r + 1; k < 6; ++k) s -= A[r][k] * xi[k];
    xi[r] = s / A[r][r];
  }

  // se3_exp(xi)
  const float vx = xi[0], vy = xi[1], vz = xi[2];
  const float wx = xi[3], wy = xi[4], wz = xi[5];
  const float t2 = wx * wx + wy * wy + wz * wz;
  float Ac, Bc, Cc;
  if (t2 < 1e-8f) {
    Ac = 1.f - t2 / 6.f; Bc = 0.5f - t2 / 24.f; Cc = 1.f / 6.f - t2 / 120.f;
  } else {
    const float th = sqrtf(t2), sn = sinf(th), cs = cosf(th);
    Ac = sn / th; Bc = (1.f - cs) / t2; Cc = (th - sn) / (t2 * th);
  }
  const float Wm[3][3] = { {0.f, -wz, wy}, {wz, 0.f, -wx}, {-wy, wx, 0.f} };
  float W2[3][3];
  for (int i = 0; i < 3; ++i)
    for (int j = 0; j < 3; ++j) {
      float s = 0.f;
      for (int k = 0; k < 3; ++k) s += Wm[i][k] * Wm[k][j];
      W2[i][j] = s;
    }
  float T[4][4];
  for (int i = 0; i < 3; ++i)
    for (int j = 0; j < 3; ++j)
      T[i][j] = (i == j ? 1.f : 0.f) + Ac * Wm[i][j] + Bc * W2[i][j];
  float V[3][3];
  for (int i = 0; i < 3; ++i)
    for (int j = 0; j < 3; ++j)
      V[i][j] = (i == j ? 1.f : 0.f) + Bc * Wm[i][j] + Cc * W2[i][j];
  T[0][3] = V[0][0] * vx + V[0][1] * vy + V[0][2] * vz;
  T[1][3] = V[1][0] * vx + V[1][1] * vy + V[1][2] * vz;
  T[2][3] = V[2][0] * vx + V[2][1] * vy + V[2][2] * vz;
  T[3][0] = 0.f; T[3][1] = 0.f; T[3][2] = 0.f; T[3][3] = 1.f;

  // pose <- T * pose
  float Po[16], Pn[16];
  for (int i = 0; i < 16; ++i) Po[i] = pose[b * 16 + i];
  for (int i = 0; i < 4; ++i)
    for (int j = 0; j < 4; ++j) {
      float s = 0.f;
      for (int k = 0; k < 4; ++k) s += T[i][k] * Po[k * 4 + j];
      Pn[i * 4 + j] = s;
    }
  for (int i = 0; i < 16; ++i) pose[b * 16 + i] = Pn[i];
}

// ---------------------------------------------------------------------------
// Inputs (setup_inputs order): 0 pose, 1 I0, 2 I1, 3 invD0, 4 invD1,
// 5 intrinsics, 6 depth0, 7 depth1.  Output: final pose (B x 4 x 4).
// ---------------------------------------------------------------------------
extern "C" void kernel_launch(void* const* d_in, const int* in_sizes, int n_in,
                              void* d_out, int out_size, void* d_ws, size_t ws_size,
                              hipStream_t stream) {
  (void)in_sizes; (void)n_in; (void)out_size; (void)ws_size;
  const float* pose_in = (const float*)d_in[0];
  const float* I0      = (const float*)d_in[1];
  const float* I1      = (const float*)d_in[2];
  const float* intr    = (const float*)d_in[5];
  const float* depth0  = (const float*)d_in[6];
  const float* depth1  = (const float*)d_in[7];
  float* pose    = (float*)d_out;
  float* partial = (float*)d_ws;   // kB * kNB * 49 floats

  hipMemcpyAsync(pose, pose_in, (size_t)kB * 16 * sizeof(float),
                 hipMemcpyDeviceToDevice, stream);
  for (int it = 0; it < kNIter; ++it) {
    dia_reduce<<<dim3(kB * kNB), dim3(kBlock), 0, stream>>>(
        pose, I0, I1, intr, depth0, depth1, partial);
    dia_solve<<<dim3(kB), dim3(64), 0, stream>>>(partial, pose);
  }
}